// AIGDiscriminator_55482387530049
// MI455X (gfx1250) — compile-verified
//
#include <hip/hip_runtime.h>
#include <math.h>

#define N_NODES 500000
#define BATCH   64
#define LVL     512
#define HID     64
#define CCONV   64

typedef __attribute__((ext_vector_type(2))) float v2f;
typedef __attribute__((ext_vector_type(8))) float v8f;

// ---------------- degree / normalization ----------------
__global__ void k_deg(const int* __restrict__ ei, float* __restrict__ deg, int E) {
    int t = blockIdx.x * blockDim.x + threadIdx.x;
    if (t < E) atomicAdd(&deg[ei[E + t]], 1.0f);   // dst row of edge_index
}

__global__ void k_dinv(float* __restrict__ deg, int n) {
    int t = blockIdx.x * blockDim.x + threadIdx.x;
    if (t < n) deg[t] = rsqrtf(deg[t] + 1.0f);     // +1 self loop
}

// ---------------- layer1 projection: x[N,2] @ W1[2,64] ----------------
__global__ void k_xw1(const float* __restrict__ x, const float* __restrict__ W1,
                      float* __restrict__ xw, int total) {
    int t = blockIdx.x * blockDim.x + threadIdx.x;
    if (t < total) {
        int i = t >> 6, j = t & 63;
        xw[t] = x[2 * i] * W1[j] + x[2 * i + 1] * W1[64 + j];
    }
}

// ---------------- edge scatter-add: agg[dst] += norm * xw[src] ----------------
__global__ void k_edge_agg(const int* __restrict__ ei, const float* __restrict__ dinv,
                           const float* __restrict__ xw, float* __restrict__ agg, int E) {
    int t = blockIdx.x * blockDim.x + threadIdx.x;
    if (t < E * 64) {
        int e = t >> 6, j = t & 63;
        int s = ei[e], d = ei[E + e];
        float norm = dinv[s] * dinv[d];
        atomicAdd(&agg[d * 64 + j], norm * xw[s * 64 + j]);
    }
}

// ---------------- add self-loop term + bias + relu (in place on agg) ----------------
__global__ void k_finish(const float* __restrict__ xw, const float* __restrict__ dinv,
                         const float* __restrict__ bias, float* __restrict__ agg, int total) {
    int t = blockIdx.x * blockDim.x + threadIdx.x;
    if (t < total) {
        int i = t >> 6, j = t & 63;
        float di = dinv[i];
        float v = agg[t] + di * di * xw[t] + bias[j];
        agg[t] = v > 0.0f ? v : 0.0f;
    }
}

// ---------------- WMMA fp32 GEMM: D[rows,64] = A[rows,64] @ W[64,64] ----------------
// One wave computes a 16x64 output tile: 16 K-steps (K=4 each) x 4 N-tiles.
__global__ __launch_bounds__(128)
void k_gemm_wmma(const float* __restrict__ A, const float* __restrict__ W,
                 float* __restrict__ D, int rows) {
    int lane = threadIdx.x & 31;
    int wave = threadIdx.x >> 5;
    int tile = blockIdx.x * 4 + wave;
    if (tile * 16 >= rows) return;            // wave-uniform guard: EXEC stays all-ones
    int row0 = tile * 16;

    int m     = lane & 15;                    // M (A) / N (B,D) index within half
    int khalf = (lane >> 4) * 2;              // lanes 16-31 hold K+2,K+3

    int arow = row0 + m;
    if (arow > rows - 1) arow = rows - 1;     // safe clamp (rows divisible by 16 here)
    const float* Arow = A + (size_t)arow * 64;

    v8f acc0 = {}, acc1 = {}, acc2 = {}, acc3 = {};
    for (int k = 0; k < 64; k += 4) {
        // A 16x4 tile: VGPR0 = K+khalf, VGPR1 = K+khalf+1
        v2f a;
        a.x = Arow[k + khalf];
        a.y = Arow[k + khalf + 1];
        // B 4x16 tiles: lane n holds W[k+khalf][16c+n], W[k+khalf+1][16c+n]
        const float* Wk = W + (size_t)(k + khalf) * 64;
        v2f b0, b1, b2, b3;
        b0.x = Wk[m];           b0.y = Wk[64 + m];
        b1.x = Wk[16 + m];      b1.y = Wk[64 + 16 + m];
        b2.x = Wk[32 + m];      b2.y = Wk[64 + 32 + m];
        b3.x = Wk[48 + m];      b3.y = Wk[64 + 48 + m];
        acc0 = __builtin_amdgcn_wmma_f32_16x16x4_f32(false, a, false, b0, (short)0, acc0, false, false);
        acc1 = __builtin_amdgcn_wmma_f32_16x16x4_f32(false, a, false, b1, (short)0, acc1, false, false);
        acc2 = __builtin_amdgcn_wmma_f32_16x16x4_f32(false, a, false, b2, (short)0, acc2, false, false);
        acc3 = __builtin_amdgcn_wmma_f32_16x16x4_f32(false, a, false, b3, (short)0, acc3, false, false);
    }

    // D 16x16 fp32 layout: VGPR r -> M=r (lanes 0-15) / M=r+8 (lanes 16-31), N=lane&15
    int col    = lane & 15;
    int rowadd = (lane >> 4) * 8;
    for (int r = 0; r < 8; ++r) {
        int row = row0 + r + rowadd;
        if (row < rows) {
            float* out = D + (size_t)row * 64;
            out[col]      = acc0[r];
            out[16 + col] = acc1[r];
            out[32 + col] = acc2[r];
            out[48 + col] = acc3[r];
        }
    }
}

// ---------------- level-wise pooling: segment sum / count / max ----------------
__global__ void k_pool(const float* __restrict__ h, const int* __restrict__ depth,
                       const int* __restrict__ bidx, float* __restrict__ cnt,
                       float* __restrict__ sum, float* __restrict__ mx, int total) {
    int t = blockIdx.x * blockDim.x + threadIdx.x;
    if (t < total) {
        int i = t >> 6, j = t & 63;
        int seg = bidx[i] * LVL + depth[i];
        float v = h[t];
        atomicAdd(&sum[seg * 64 + j], v);
        // v >= 0 (post-ReLU): int bit-pattern max == float max; buffer init 0
        atomicMax((int*)&mx[seg * 64 + j], __float_as_int(v));
        if (j == 0) atomicAdd(&cnt[seg], 1.0f);
    }
}

// ---------------- build seq[b][ch][l]  (ch<64: mean, ch>=64: max) ----------------
__global__ void k_seq(const float* __restrict__ cnt, const float* __restrict__ sum,
                      const float* __restrict__ mx, float* __restrict__ seq, int total) {
    int t = blockIdx.x * blockDim.x + threadIdx.x;
    if (t < total) {
        int l  = t & (LVL - 1);
        int ch = (t >> 9) & 127;
        int b  = t >> 16;                 // 512*128 = 65536
        int seg = b * LVL + l;
        float v;
        if (ch < 64) {
            float c = cnt[seg];
            v = sum[seg * 64 + ch] / fmaxf(c, 1.0f);
        } else {
            v = mx[seg * 64 + (ch - 64)];
        }
        seq[t] = v;
    }
}

// ---------------- conv1d (K=3, pad 1) + relu + mean over L ----------------
__global__ __launch_bounds__(128)
void k_conv(const float* __restrict__ seq, const float* __restrict__ w,
            const float* __restrict__ cb, float* __restrict__ pooled) {
    int b = blockIdx.x >> 6;
    int o = blockIdx.x & 63;
    const float* wo = w + o * 128 * 3;
    const float* sb = seq + (size_t)b * 128 * LVL;
    float bias = cb[o];
    float total = 0.0f;
    for (int l = threadIdx.x; l < LVL; l += 128) {
        float acc = bias;
        for (int ci = 0; ci < 128; ++ci) {
            const float* s  = sb + ci * LVL + l;
            const float* wi = wo + ci * 3;
            if (l > 0)       acc += s[-1] * wi[0];
            acc += s[0] * wi[1];
            if (l < LVL - 1) acc += s[1] * wi[2];
        }
        total += fmaxf(acc, 0.0f);
    }
    __shared__ float red[128];
    red[threadIdx.x] = total;
    __syncthreads();
    for (int s = 64; s > 0; s >>= 1) {
        if (threadIdx.x < s) red[threadIdx.x] += red[threadIdx.x + s];
        __syncthreads();
    }
    if (threadIdx.x == 0) pooled[b * 64 + o] = red[0] / (float)LVL;
}

// ---------------- MLP head + sigmoid ----------------
__global__ void k_head(const float* __restrict__ pooled, const float* __restrict__ fc1w,
                       const float* __restrict__ fc1b, const float* __restrict__ fc2w,
                       const float* __restrict__ fc2b, float* __restrict__ out) {
    int b = threadIdx.x;
    if (b < BATCH) {
        const float* p = pooled + b * 64;
        float logit = fc2b[0];
        for (int k = 0; k < 64; ++k) {
            float z = fc1b[k];
            for (int c = 0; c < 64; ++c) z += p[c] * fc1w[c * 64 + k];
            z = fmaxf(z, 0.0f);
            logit += z * fc2w[k];
        }
        out[b] = 1.0f / (1.0f + expf(-logit));
    }
}

extern "C" void kernel_launch(void* const* d_in, const int* in_sizes, int n_in,
                              void* d_out, int out_size, void* d_ws, size_t ws_size,
                              hipStream_t stream) {
    const float* x     = (const float*)d_in[0];
    const int*   ei    = (const int*)d_in[1];
    const int*   depth = (const int*)d_in[2];
    const int*   bidx  = (const int*)d_in[3];
    const float* W1    = (const float*)d_in[4];
    const float* b1    = (const float*)d_in[5];
    const float* W2    = (const float*)d_in[6];
    const float* b2    = (const float*)d_in[7];
    const float* convw = (const float*)d_in[8];
    const float* convb = (const float*)d_in[9];
    const float* fc1w  = (const float*)d_in[10];
    const float* fc1b  = (const float*)d_in[11];
    const float* fc2w  = (const float*)d_in[12];
    const float* fc2b  = (const float*)d_in[13];

    const int E  = in_sizes[1] / 2;
    const int Nn = in_sizes[2];

    // workspace layout (256B aligned)
    char* ws = (char*)d_ws;
    size_t off = 0;
    auto alloc = [&](size_t bytes) -> void* {
        void* p = ws + off;
        off = (off + bytes + 255) & ~(size_t)255;
        return p;
    };
    float* bufA   = (float*)alloc((size_t)Nn * 64 * 4);        // xw
    float* bufB   = (float*)alloc((size_t)Nn * 64 * 4);        // agg / h
    float* dinv   = (float*)alloc((size_t)Nn * 4);
    float* cnt    = (float*)alloc((size_t)BATCH * LVL * 4);
    float* sum    = (float*)alloc((size_t)BATCH * LVL * 64 * 4);
    float* mx     = (float*)alloc((size_t)BATCH * LVL * 64 * 4);
    float* seq    = (float*)alloc((size_t)BATCH * 128 * LVL * 4);
    float* pooled = (float*)alloc((size_t)BATCH * 64 * 4);

    const int NT = Nn * 64;
    const int ET = E * 64;

    // degrees -> dinv
    hipMemsetAsync(dinv, 0, (size_t)Nn * 4, stream);
    k_deg<<<(E + 255) / 256, 256, 0, stream>>>(ei, dinv, E);
    k_dinv<<<(Nn + 255) / 256, 256, 0, stream>>>(dinv, Nn);

    // ---- GCN layer 1 ----
    k_xw1<<<(NT + 255) / 256, 256, 0, stream>>>(x, W1, bufA, NT);
    hipMemsetAsync(bufB, 0, (size_t)Nn * 64 * 4, stream);
    k_edge_agg<<<(ET + 255) / 256, 256, 0, stream>>>(ei, dinv, bufA, bufB, E);
    k_finish<<<(NT + 255) / 256, 256, 0, stream>>>(bufA, dinv, b1, bufB, NT);   // bufB = h1

    // ---- GCN layer 2: xw2 = h1 @ W2 (WMMA fp32), then aggregate ----
    int tiles = (Nn + 15) / 16;
    k_gemm_wmma<<<(tiles + 3) / 4, 128, 0, stream>>>(bufB, W2, bufA, Nn);       // bufA = xw2
    hipMemsetAsync(bufB, 0, (size_t)Nn * 64 * 4, stream);
    k_edge_agg<<<(ET + 255) / 256, 256, 0, stream>>>(ei, dinv, bufA, bufB, E);
    k_finish<<<(NT + 255) / 256, 256, 0, stream>>>(bufA, dinv, b2, bufB, NT);   // bufB = h2

    // ---- level-wise pooling ----
    hipMemsetAsync(cnt, 0, (size_t)BATCH * LVL * 4, stream);
    hipMemsetAsync(sum, 0, (size_t)BATCH * LVL * 64 * 4, stream);
    hipMemsetAsync(mx, 0, (size_t)BATCH * LVL * 64 * 4, stream);
    k_pool<<<(NT + 255) / 256, 256, 0, stream>>>(bufB, depth, bidx, cnt, sum, mx, NT);

    int ST = BATCH * 128 * LVL;
    k_seq<<<(ST + 255) / 256, 256, 0, stream>>>(cnt, sum, mx, seq, ST);

    // ---- conv + mean pool + head ----
    k_conv<<<BATCH * 64, 128, 0, stream>>>(seq, convw, convb, pooled);
    k_head<<<1, 64, 0, stream>>>(pooled, fc1w, fc1b, fc2w, fc2b, (float*)d_out);
}